// DisplaceChannel_70652212019726
// MI455X (gfx1250) — compile-verified
//
#include <hip/hip_runtime.h>

// Problem constants (fixed by the reference)
#define HH 128
#define WW 128
#define BB 16
#define PP 48
#define CPP 4
#define CC (PP * CPP)   // 192

#define TILE_H 32
#define LDSH   (TILE_H + 2)   // 34 rows (1 halo row each side)
#define LDSW   132            // 130 used cols (+pad) ; 132 % 64 == 4 avoids systematic conflicts
#define NT     256            // 8 wave32

#if defined(__has_builtin)
# if __has_builtin(__builtin_amdgcn_global_load_async_to_lds_b32)
#  define HAVE_ASYNC_LDS 1
# endif
# if __has_builtin(__builtin_amdgcn_s_wait_asynccnt)
#  define HAVE_WAIT_ASYNC 1
# endif
#endif

typedef __attribute__((address_space(1))) int as1_int;   // global
typedef __attribute__((address_space(3))) int as3_int;   // LDS

__global__ __launch_bounds__(NT) void displace_lo_fused(
    const float* __restrict__ inp,      // (B, C, H, W)
    const float* __restrict__ offset,   // (P, 2) [x, y]
    float* __restrict__ out)            // (B, C, H, W)
{
    const int tid = threadIdx.x;
    const int y0  = blockIdx.x * TILE_H;   // tile start row
    const int c   = blockIdx.y;            // channel
    const int b   = blockIdx.z;            // batch
    const int p   = c >> 2;                // position = c / CPP

    // ---- per-position separable Gaussian weights (uniform across block) ----
    const float ox = offset[2 * p + 0];
    const float oy = offset[2 * p + 1];
    const float rx = rintf(ox);            // round-half-even == jnp.round
    const float ry = rintf(oy);
    const int   Rx = (int)rx;
    const int   Ry = (int)ry;
    const float dx = ox - rx;
    const float dy = oy - ry;

    float wx[3], wy[3];
    float sx = 0.0f, sy = 0.0f;
    #pragma unroll
    for (int k = 0; k < 3; ++k) {
        const float fx = (float)(k - 1) + dx;
        const float fy = (float)(k - 1) + dy;
        wx[k] = expf(-2.0f * fx * fx);     // 1/(2*sigma^2) = 2.0 for sigma = 0.5
        wy[k] = expf(-2.0f * fy * fy);
        sx += wx[k];
        sy += wy[k];
    }
    #pragma unroll
    for (int k = 0; k < 3; ++k) { wx[k] /= sx; wy[k] /= sy; }  // kern = (wy ⊗ wx), Σ = 1

    // ---- LDS tile of the *shifted* image S with 1-px halo ----
    // S(yy,xx) = inp(yy-Ry, xx-Rx) if in-bounds else 0 ; stored at s[yy-(y0-1)][xx+1]
    __shared__ float s[LDSH * LDSW];

    for (int i = tid; i < LDSH * LDSW; i += NT) s[i] = 0.0f;   // covers shift-OOB + conv pad
    __syncthreads();                                            // ds stores done before async writes

    // Valid xx columns form one contiguous run: [max(0,Rx), min(W, W+Rx))
    const int xx_s = (Rx > 0) ? Rx : 0;
    const int xx_e = (Rx < 0) ? (WW + Rx) : WW;
    const size_t planeBase = ((size_t)b * CC + c) * (size_t)(HH * WW);

    for (int e = tid; e < LDSH * WW; e += NT) {        // 34*128/256 = 17 iters
        const int j  = e >> 7;                         // LDS row
        const int t  = e & (WW - 1);
        const int xx = xx_s + t;
        const int yy = y0 - 1 + j;                     // conv-space row
        const int ys = yy - Ry;                        // source row
        if (xx < xx_e && yy >= 0 && yy < HH && ys >= 0 && ys < HH) {
            const int xs = xx - Rx;                    // source col (contiguous per lane)
            const float* g = inp + planeBase + (size_t)ys * WW + xs;
            float*       l = &s[j * LDSW + xx + 1];
#ifdef HAVE_ASYNC_LDS
            __builtin_amdgcn_global_load_async_to_lds_b32(
                (as1_int*)g, (as3_int*)l, /*imm offset*/ 0, /*cpol*/ 0);
#else
            *l = *g;
#endif
        }
    }
#ifdef HAVE_ASYNC_LDS
# ifdef HAVE_WAIT_ASYNC
    __builtin_amdgcn_s_wait_asynccnt(0);   // ASYNCcnt not covered by __syncthreads' waits
# else
    asm volatile("s_wait_asynccnt 0" ::: "memory");
# endif
#endif
    __syncthreads();

    // ---- separable 3x3: vertical blend of 6 cols, then horizontal; 4 rows x 4 cols per thread
    const int r0  = tid >> 5;          // 0..7
    const int x0c = (tid & 31) * 4;    // 0..124 (16B-aligned stores)

    #pragma unroll
    for (int h = 0; h < TILE_H / 8; ++h) {
        const int r = r0 + 8 * h;          // tile row 0..31
        const int y = y0 + r;
        const float* row0 = &s[(r + 0) * LDSW + x0c];  // S row y-1
        const float* row1 = &s[(r + 1) * LDSW + x0c];  // S row y
        const float* row2 = &s[(r + 2) * LDSW + x0c];  // S row y+1

        float v[6];
        #pragma unroll
        for (int k = 0; k < 6; ++k)
            v[k] = wy[0] * row0[k] + wy[1] * row1[k] + wy[2] * row2[k];

        float4 o;
        o.x = wx[0] * v[0] + wx[1] * v[1] + wx[2] * v[2];
        o.y = wx[0] * v[1] + wx[1] * v[2] + wx[2] * v[3];
        o.z = wx[0] * v[2] + wx[1] * v[3] + wx[2] * v[4];
        o.w = wx[0] * v[3] + wx[1] * v[4] + wx[2] * v[5];

        *(float4*)(out + planeBase + (size_t)y * WW + x0c) = o;
    }
}

extern "C" void kernel_launch(void* const* d_in, const int* in_sizes, int n_in,
                              void* d_out, int out_size, void* d_ws, size_t ws_size,
                              hipStream_t stream) {
    const float* inp    = (const float*)d_in[0];   // (16, 192, 128, 128) f32
    const float* offset = (const float*)d_in[1];   // (48, 2) f32
    float*       out    = (float*)d_out;           // (16, 192, 128, 128) f32

    dim3 grid(HH / TILE_H, CC, BB);   // 4 x 192 x 16
    displace_lo_fused<<<grid, NT, 0, stream>>>(inp, offset, out);
}